// RIM_GRU_44289702756725
// MI455X (gfx1250) — compile-verified
//
#include <hip/hip_runtime.h>
#include <hip/hip_bf16.h>
#include <math.h>

// ---------------- problem constants ----------------
#define TT      64
#define BB      128
#define NINP    1024
#define NHID    2048
#define NBLK    16
#define TOPK    8
#define BS      128          // NHID / NBLK
#define ATT_OUT 512          // 4*BS
#define DK1     64
#define G3      384          // 3*BS
#define NH2     4
#define DK2     16
#define DV2     16

typedef __attribute__((ext_vector_type(2))) float v2f;
typedef __attribute__((ext_vector_type(8))) float v8f;

// fp32 WMMA: D(16x16) = A(16x4) * B(4x16) + C, exact fp32 MACs.
#define WMMA_F32X4(a, b, c) \
  __builtin_amdgcn_wmma_f32_16x16x4_f32(false, (a), false, (b), (short)0, (c), false, false)

// ---------------------------------------------------------------------------
// Prep: pack B operands into WMMA-pair layout  Bp[(kp*N + n)*2 + i] = B[2kp+i][n]
// so each lane's (B[kk][col], B[kk+1][col]) pair is one contiguous b64 load.
// ---------------------------------------------------------------------------
__global__ void pack_b_kernel(const float* __restrict__ src,   // (K, N) row-major
                              float* __restrict__ dst,         // packed
                              int K, int N) {
  int idx = blockIdx.x * 256 + threadIdx.x;
  if (idx >= K * N) return;
  int n = idx % N;
  int k = idx / N;
  dst[((size_t)(k >> 1) * N + n) * 2 + (k & 1)] = src[idx];
}

// Wih (16,384,512): B matrix per block is (K=512 rows=d, N=384 cols=g), packed.
__global__ void pack_wih_kernel(const float* __restrict__ Wih,
                                float* __restrict__ WihP) {
  int idx = blockIdx.x * 256 + threadIdx.x;          // 16*384*512
  if (idx >= NBLK * G3 * ATT_OUT) return;
  int d = idx % ATT_OUT;
  int g = (idx / ATT_OUT) % G3;
  int k = idx / (ATT_OUT * G3);
  WihP[(size_t)k * (ATT_OUT * G3) + ((size_t)(d >> 1) * G3 + g) * 2 + (d & 1)] = Wih[idx];
}

// Whh (16,384,128): B matrix per block is (K=128 rows=d, N=384 cols=g), packed.
__global__ void pack_whh_kernel(const float* __restrict__ Whh,
                                float* __restrict__ WhhP) {
  int idx = blockIdx.x * 256 + threadIdx.x;          // 16*384*128
  if (idx >= NBLK * G3 * BS) return;
  int d = idx % BS;
  int g = (idx / BS) % G3;
  int k = idx / (BS * G3);
  WhhP[(size_t)k * (BS * G3) + ((size_t)(d >> 1) * G3 + g) * 2 + (d & 1)] = Whh[idx];
}

// ---------------------------------------------------------------------------
// One-time hoisted GEMM: K_all = x@Wk1+bk1 (8192x64), V_all = x@Wv1+bv1 (8192x512).
// Each wave: one N-tile x 4 M-tiles (B fragment amortized 4x). grid = 576 WGs.
// ---------------------------------------------------------------------------
__global__ void kv_gemm_kernel(const float* __restrict__ x,
                               const float* __restrict__ Wk1P, const float* __restrict__ bk1,
                               const float* __restrict__ Wv1P, const float* __restrict__ bv1,
                               float* __restrict__ K_all, float* __restrict__ V_all) {
  const int wave = threadIdx.x >> 5;
  const int lane = threadIdx.x & 31;
  const int lm   = lane & 15;
  const int kh   = lane >> 4;                         // 0 or 1

  int unit   = blockIdx.x * 8 + wave;                 // 0 .. 4607
  int ntile  = unit % 36;
  int mgroup = unit / 36;                             // 0 .. 127 (64 rows each)
  int m0     = mgroup * 64;

  const float* Bp;  const float* bias;  float* Out;  int N, col0;
  if (ntile < 4) { Bp = Wk1P; bias = bk1; Out = K_all; N = DK1;     col0 = ntile * 16; }
  else           { Bp = Wv1P; bias = bv1; Out = V_all; N = ATT_OUT; col0 = (ntile - 4) * 16; }
  const int col = col0 + lm;

  v8f acc[4] = {};
  for (int k0 = 0; k0 < NINP; k0 += 4) {
    int kk = k0 + 2 * kh;
    int kp = (k0 >> 1) + kh;
    v2f b = *(const v2f*)(Bp + ((size_t)kp * N + col) * 2);
#pragma unroll
    for (int mt = 0; mt < 4; ++mt) {
      const float* Arow = x + (size_t)(m0 + mt * 16 + lm) * NINP;
      v2f a = *(const v2f*)(Arow + kk);
      acc[mt] = WMMA_F32X4(a, b, acc[mt]);
    }
  }
  float bb = bias[col];
#pragma unroll
  for (int mt = 0; mt < 4; ++mt)
#pragma unroll
    for (int r = 0; r < 8; ++r)
      Out[(size_t)(m0 + mt * 16 + r + 8 * kh) * N + col] = acc[mt][r] + bb;
}

// ---------------------------------------------------------------------------
// Per step: q = h_blk@Wq1+bq1; 2-key softmax; inp_use = p0*bv1 + p1*V_t.
// One wave per (b, block).
// ---------------------------------------------------------------------------
__global__ void att1_kernel(const float* __restrict__ hprev,
                            const float* __restrict__ Wq1, const float* __restrict__ bq1,
                            const float* __restrict__ bk1, const float* __restrict__ bv1,
                            const float* __restrict__ Kt,   // K_all + t*B*DK1
                            const float* __restrict__ Vt,   // V_all + t*B*ATT_OUT
                            float* __restrict__ inp_use, float* __restrict__ null_sc) {
  const int wave = threadIdx.x >> 5;
  const int lane = threadIdx.x & 31;
  const int idx  = blockIdx.x * 8 + wave;             // 0 .. B*NBLK-1
  const int b    = idx >> 4;
  const int k    = idx & 15;

  const float* h = hprev + (size_t)b * NHID + k * BS;
  float q0 = bq1[lane], q1 = bq1[lane + 32];
  for (int d = 0; d < BS; ++d) {
    float hv = h[d];
    q0 += hv * Wq1[d * DK1 + lane];
    q1 += hv * Wq1[d * DK1 + lane + 32];
  }
  const float* kt = Kt + (size_t)b * DK1;
  float s0 = q0 * bk1[lane] + q1 * bk1[lane + 32];
  float s1 = q0 * kt[lane]  + q1 * kt[lane + 32];
  for (int off = 16; off >= 1; off >>= 1) {
    s0 += __shfl_xor(s0, off, 32);
    s1 += __shfl_xor(s1, off, 32);
  }
  const float inv_t1 = 0.125f;                        // 1/sqrt(DK1)
  s0 *= inv_t1; s1 *= inv_t1;
  float mx = fmaxf(s0, s1);
  float e0 = expf(s0 - mx), e1 = expf(s1 - mx);
  float den = e0 + e1;
  float p0 = e0 / den, p1 = e1 / den;
  if (lane == 0) null_sc[idx] = p0;

  const float* vt = Vt + (size_t)b * ATT_OUT;
  float* iu = inp_use + (size_t)idx * ATT_OUT;
#pragma unroll
  for (int i = 0; i < 16; ++i) {
    int d = lane + 32 * i;
    iu[d] = p0 * bv1[d] + p1 * vt[d];
  }
}

// ---------------------------------------------------------------------------
// Per step: drop the (NB-TOPK) blocks with largest null score (stable ties).
// ---------------------------------------------------------------------------
__global__ void mask_kernel(const float* __restrict__ null_sc,
                            float* __restrict__ maskb) {
  int b = threadIdx.x;                                // block = 128, grid = 1
  float ns[NBLK];
#pragma unroll
  for (int k = 0; k < NBLK; ++k) ns[k] = null_sc[b * NBLK + k];
#pragma unroll
  for (int k = 0; k < NBLK; ++k) {
    int rank = 0;
    for (int j = 0; j < NBLK; ++j)
      if (ns[j] > ns[k] || (ns[j] == ns[k] && j < k)) rank++;
    maskb[b * NBLK + k] = (rank < (NBLK - TOPK)) ? 0.0f : 1.0f;
  }
}

// ---------------------------------------------------------------------------
// Per step (dominant GEMM): per-block gates.
//   acc_i over K=512 (inp_use x WihP), acc_h over K=128 (h x WhhP).
//   N-tiles 0..15 -> combined r/z pre-activations; 16..23 -> gi_n, gh_n apart.
// grid = (NBLK*3, 2); 8 waves/WG, each wave: 1 N-tile x 4 M-tiles.
// ---------------------------------------------------------------------------
__global__ void gates_gemm_kernel(const float* __restrict__ inp_use,
                                  const float* __restrict__ hprev,
                                  const float* __restrict__ WihP,
                                  const float* __restrict__ WhhP,
                                  const float* __restrict__ bih,
                                  const float* __restrict__ bhh,
                                  float* __restrict__ gates_rz,
                                  float* __restrict__ gin,
                                  float* __restrict__ ghn) {
  const int wave = threadIdx.x >> 5;
  const int lane = threadIdx.x & 31;
  const int lm   = lane & 15;
  const int kh   = lane >> 4;

  const int kblk   = blockIdx.x / 3;
  const int ngroup = blockIdx.x % 3;
  const int ntile  = ngroup * 8 + wave;               // 0..23
  const int col0   = ntile * 16;
  const int col    = col0 + lm;
  const int mbase  = blockIdx.y * 64;                 // 4 M-tiles of 16 batch rows

  const float* Bi = WihP + (size_t)kblk * (ATT_OUT * G3);
  const float* Bh = WhhP + (size_t)kblk * (BS * G3);

  v8f acc_i[4] = {};
  v8f acc_h[4] = {};

  for (int k0 = 0; k0 < ATT_OUT; k0 += 4) {
    int kk = k0 + 2 * kh;
    int kp = (k0 >> 1) + kh;
    v2f b = *(const v2f*)(Bi + ((size_t)kp * G3 + col) * 2);
#pragma unroll
    for (int mt = 0; mt < 4; ++mt) {
      const float* Ai = inp_use + ((size_t)(mbase + mt * 16 + lm) * NBLK + kblk) * ATT_OUT;
      v2f a = *(const v2f*)(Ai + kk);
      acc_i[mt] = WMMA_F32X4(a, b, acc_i[mt]);
    }
  }
  for (int k0 = 0; k0 < BS; k0 += 4) {
    int kk = k0 + 2 * kh;
    int kp = (k0 >> 1) + kh;
    v2f b = *(const v2f*)(Bh + ((size_t)kp * G3 + col) * 2);
#pragma unroll
    for (int mt = 0; mt < 4; ++mt) {
      const float* Ah = hprev + (size_t)(mbase + mt * 16 + lm) * NHID + kblk * BS;
      v2f a = *(const v2f*)(Ah + kk);
      acc_h[mt] = WMMA_F32X4(a, b, acc_h[mt]);
    }
  }

  const float bi = bih[kblk * G3 + col];
  const float bh = bhh[kblk * G3 + col];
#pragma unroll
  for (int mt = 0; mt < 4; ++mt) {
#pragma unroll
    for (int r = 0; r < 8; ++r) {
      int mrow = mbase + mt * 16 + r + 8 * kh;
      if (ntile < 16) {
        gates_rz[((size_t)mrow * NBLK + kblk) * (2 * BS) + col] =
            acc_i[mt][r] + acc_h[mt][r] + bi + bh;
      } else {
        int j = col - 2 * BS;                         // 0..127
        gin[((size_t)mrow * NBLK + kblk) * BS + j] = acc_i[mt][r] + bi;
        ghn[((size_t)mrow * NBLK + kblk) * BS + j] = acc_h[mt][r] + bh;
      }
    }
  }
}

// ---------------------------------------------------------------------------
// Per step: GRU elementwise  hnew = (1-z)*n + z*h,  n = tanh(gi_n + r*gh_n).
// ---------------------------------------------------------------------------
__global__ void gru_elem_kernel(const float* __restrict__ gates_rz,
                                const float* __restrict__ gin,
                                const float* __restrict__ ghn,
                                const float* __restrict__ hprev,
                                float* __restrict__ hnew) {
  int idx = blockIdx.x * 256 + threadIdx.x;           // B*NHID
  if (idx >= BB * NHID) return;
  int j  = idx & (BS - 1);
  int bk = idx >> 7;                                  // b*NBLK + k
  float r = 1.0f / (1.0f + expf(-gates_rz[(size_t)bk * (2 * BS) + j]));
  float z = 1.0f / (1.0f + expf(-gates_rz[(size_t)bk * (2 * BS) + BS + j]));
  float n = tanhf(gin[idx] + r * ghn[idx]);
  float hp = hprev[idx];
  hnew[idx] = (1.0f - z) * n + z * hp;
}

// ---------------------------------------------------------------------------
// Per step: att2 (4 heads over the 16 blocks), gated FC, masked state update.
// One 256-thread workgroup per batch element.
// ---------------------------------------------------------------------------
__global__ void att2_kernel(const float* __restrict__ hnew,
                            const float* __restrict__ hprev,
                            const float* __restrict__ maskb,
                            const float* __restrict__ Wq2, const float* __restrict__ bq2,
                            const float* __restrict__ Wk2, const float* __restrict__ bk2,
                            const float* __restrict__ Wv2, const float* __restrict__ bv2,
                            const float* __restrict__ Wfc2, const float* __restrict__ bfc2,
                            const float* __restrict__ Wg2,  const float* __restrict__ bg2,
                            float* __restrict__ out_t) {
  __shared__ float s_h[NBLK * BS];
  __shared__ float s_q[NBLK * 64];
  __shared__ float s_k[NBLK * 64];
  __shared__ float s_v[NBLK * 64];
  __shared__ float s_o[NBLK * 64];

  const int b   = blockIdx.x;
  const int tid = threadIdx.x;
  const float* hb = hnew + (size_t)b * NHID;

  for (int i = tid; i < NHID; i += 256) s_h[i] = hb[i];
  __syncthreads();

  // q2/k2/v2 : 3 * 16 * 64 outputs (K = 128)
  for (int i = tid; i < 3 * NBLK * 64; i += 256) {
    int which = i / (NBLK * 64);
    int rem   = i % (NBLK * 64);
    int blk   = rem >> 6;
    int c     = rem & 63;
    const float* W    = (which == 0) ? Wq2 : (which == 1) ? Wk2 : Wv2;
    const float* bias = (which == 0) ? bq2 : (which == 1) ? bk2 : bv2;
    float acc = bias[c];
    const float* hrow = s_h + blk * BS;
    for (int d = 0; d < BS; ++d) acc += hrow[d] * W[d * 64 + c];
    float* dst = (which == 0) ? s_q : (which == 1) ? s_k : s_v;
    dst[blk * 64 + c] = acc;
  }
  __syncthreads();

  // attention across blocks per head: thread = (head, n-block)
  if (tid < NH2 * NBLK) {
    int h = tid >> 4;
    int n = tid & 15;
    float sc[NBLK];
    float mx = -1e30f;
    for (int m = 0; m < NBLK; ++m) {
      float s = 0.0f;
      for (int d = 0; d < DK2; ++d)
        s += s_q[n * 64 + h * DK2 + d] * s_k[m * 64 + h * DK2 + d];
      s *= 0.25f;                                     // 1/sqrt(DK2)
      sc[m] = s;
      mx = fmaxf(mx, s);
    }
    float sum = 0.0f;
    for (int m = 0; m < NBLK; ++m) { sc[m] = expf(sc[m] - mx); sum += sc[m]; }
    float inv = 1.0f / sum;
    for (int d = 0; d < DV2; ++d) {
      float o = 0.0f;
      for (int m = 0; m < NBLK; ++m) o += sc[m] * s_v[m * 64 + h * DV2 + d];
      s_o[n * 64 + h * DV2 + d] = o * inv;
    }
  }
  __syncthreads();

  // att_out = sigmoid(o@Wg2+bg2) * tanh(o@Wfc2+bfc2); masked state update
  const float* hprev_b = hprev + (size_t)b * NHID;
  float* outp = out_t + (size_t)b * NHID;
  for (int i = tid; i < NHID; i += 256) {
    int blk = i >> 7;
    int j   = i & 127;
    float afc = bfc2[j], ag = bg2[j];
    const float* orow = s_o + blk * 64;
    for (int d = 0; d < 64; ++d) {
      float ov = orow[d];
      afc += ov * Wfc2[d * BS + j];
      ag  += ov * Wg2[d * BS + j];
    }
    float att = (1.0f / (1.0f + expf(-ag))) * tanhf(afc);
    float m   = maskb[b * NBLK + blk];
    float hn  = s_h[i] + att;
    outp[i] = m * hn + (1.0f - m) * hprev_b[i];
  }
}

// ---------------------------------------------------------------------------
// Host-side orchestration (all launches on `stream`; graph-capture safe).
// ---------------------------------------------------------------------------
extern "C" void kernel_launch(void* const* d_in, const int* in_sizes, int n_in,
                              void* d_out, int out_size, void* d_ws, size_t ws_size,
                              hipStream_t stream) {
  (void)in_sizes; (void)n_in; (void)out_size; (void)ws_size;

  const float* x    = (const float*)d_in[0];
  const float* h0   = (const float*)d_in[1];
  // d_in[2] = seq_len (unused; T fixed)
  const float* Wq1  = (const float*)d_in[3];
  const float* bq1  = (const float*)d_in[4];
  const float* Wk1  = (const float*)d_in[5];
  const float* bk1  = (const float*)d_in[6];
  const float* Wv1  = (const float*)d_in[7];
  const float* bv1  = (const float*)d_in[8];
  const float* Wih  = (const float*)d_in[9];
  const float* bih  = (const float*)d_in[10];
  const float* Whh  = (const float*)d_in[11];
  const float* bhh  = (const float*)d_in[12];
  const float* Wq2  = (const float*)d_in[13];
  const float* bq2  = (const float*)d_in[14];
  const float* Wk2  = (const float*)d_in[15];
  const float* bk2  = (const float*)d_in[16];
  const float* Wv2  = (const float*)d_in[17];
  const float* bv2  = (const float*)d_in[18];
  const float* Wfc2 = (const float*)d_in[19];
  const float* bfc2 = (const float*)d_in[20];
  const float* Wg2  = (const float*)d_in[21];
  const float* bg2  = (const float*)d_in[22];
  float* out = (float*)d_out;

  // workspace carve-out (floats)
  float* ws = (float*)d_ws;
  float* K_all    = ws; ws += (size_t)TT * BB * DK1;            // 0.5M
  float* V_all    = ws; ws += (size_t)TT * BB * ATT_OUT;        // 4.2M
  float* WihP     = ws; ws += (size_t)NBLK * ATT_OUT * G3;      // 3.1M
  float* WhhP     = ws; ws += (size_t)NBLK * BS * G3;           // 0.8M
  float* Wk1P     = ws; ws += (size_t)NINP * DK1;               // 64K
  float* Wv1P     = ws; ws += (size_t)NINP * ATT_OUT;           // 0.5M
  float* inp_use  = ws; ws += (size_t)BB * NBLK * ATT_OUT;      // 1.0M
  float* null_sc  = ws; ws += (size_t)BB * NBLK;
  float* maskb    = ws; ws += (size_t)BB * NBLK;
  float* gates_rz = ws; ws += (size_t)BB * NBLK * 2 * BS;       // 0.5M
  float* gin      = ws; ws += (size_t)BB * NBLK * BS;
  float* ghn      = ws; ws += (size_t)BB * NBLK * BS;
  float* hnew     = ws; ws += (size_t)BB * NBLK * BS;

  // one-time prep: pack all WMMA B operands, then hoisted KV GEMM
  pack_wih_kernel<<<(NBLK * G3 * ATT_OUT + 255) / 256, 256, 0, stream>>>(Wih, WihP);
  pack_whh_kernel<<<(NBLK * G3 * BS + 255) / 256, 256, 0, stream>>>(Whh, WhhP);
  pack_b_kernel<<<(NINP * DK1 + 255) / 256, 256, 0, stream>>>(Wk1, Wk1P, NINP, DK1);
  pack_b_kernel<<<(NINP * ATT_OUT + 255) / 256, 256, 0, stream>>>(Wv1, Wv1P, NINP, ATT_OUT);
  kv_gemm_kernel<<<576, 256, 0, stream>>>(x, Wk1P, bk1, Wv1P, bv1, K_all, V_all);

  // sequential scan
  for (int t = 0; t < TT; ++t) {
    const float* hprev = (t == 0) ? h0 : (out + (size_t)(t - 1) * BB * NHID);
    att1_kernel<<<BB * NBLK / 8, 256, 0, stream>>>(
        hprev, Wq1, bq1, bk1, bv1,
        K_all + (size_t)t * BB * DK1, V_all + (size_t)t * BB * ATT_OUT,
        inp_use, null_sc);
    mask_kernel<<<1, BB, 0, stream>>>(null_sc, maskb);
    gates_gemm_kernel<<<dim3(NBLK * 3, 2), 256, 0, stream>>>(
        inp_use, hprev, WihP, WhhP, bih, bhh, gates_rz, gin, ghn);
    gru_elem_kernel<<<(BB * NHID + 255) / 256, 256, 0, stream>>>(
        gates_rz, gin, ghn, hprev, hnew);
    att2_kernel<<<BB, 256, 0, stream>>>(
        hnew, hprev, maskb, Wq2, bq2, Wk2, bk2, Wv2, bv2,
        Wfc2, bfc2, Wg2, bg2, out + (size_t)t * BB * NHID);
  }
}